// DetConBLoss_69020124447445
// MI455X (gfx1250) — compile-verified
//
#include <hip/hip_runtime.h>
#include <hip/hip_bf16.h>

typedef __attribute__((ext_vector_type(2))) float v2f;
typedef __attribute__((ext_vector_type(4))) float v4f;
typedef __attribute__((ext_vector_type(8))) float v8f;

#define BATCH 256
#define NROI  16
#define DDIM  256
#define ROWS  (BATCH * NROI)          // 4096
#define TENSOR_ELEMS (ROWS * DDIM)    // 1048576 floats = 4 MB
#define LOGIT_SCALE (1.0f / (0.1f + 1e-11f))
#define LSE_BIAS 10.0f                // logits = 10*dot(unit,unit) <= 10 exactly

// ---------------------------------------------------------------------------
// Kernel 1: L2-normalize every (b,n) row of the 4 input tensors into ws.
// One wave (32 lanes) per row; each lane owns 8 contiguous floats.
// ---------------------------------------------------------------------------
__global__ __launch_bounds__(256) void detconb_normalize(
    const float* __restrict__ p1, const float* __restrict__ p2,
    const float* __restrict__ t1, const float* __restrict__ t2,
    float* __restrict__ ws)
{
    const int gw   = (blockIdx.x * blockDim.x + threadIdx.x) >> 5; // global wave id
    const int lane = threadIdx.x & 31;
    if (gw >= 4 * ROWS) return;
    const int tensor = gw >> 12;        // 0..3
    const int row    = gw & (ROWS - 1); // 0..4095

    const float* src = (tensor == 0) ? p1 : (tensor == 1) ? p2
                     : (tensor == 2) ? t1 : t2;
    const float* r  = src + (size_t)row * DDIM;
    float*       dw = ws + (size_t)tensor * TENSOR_ELEMS + (size_t)row * DDIM;

    const int off = lane * 8;
    v4f a = *(const v4f*)(r + off);
    v4f b = *(const v4f*)(r + off + 4);
    float ss = a.x*a.x + a.y*a.y + a.z*a.z + a.w*a.w
             + b.x*b.x + b.y*b.y + b.z*b.z + b.w*b.w;
    #pragma unroll
    for (int m = 1; m < 32; m <<= 1) ss += __shfl_xor(ss, m, 32);
    const float inv = 1.0f / sqrtf(ss);
    a.x *= inv; a.y *= inv; a.z *= inv; a.w *= inv;
    b.x *= inv; b.y *= inv; b.z *= inv; b.w *= inv;
    *(v4f*)(dw + off)     = a;
    *(v4f*)(dw + off + 4) = b;
}

// ---------------------------------------------------------------------------
// Kernel 2: one block per batch index b. Streams all 4096 target rows through
// fp32 WMMA (2 column tiles x 2 A-panels = 4 independent accumulator chains)
// and accumulates fixed-bias sum-exp + positive statistics per output row,
// entirely lane-local (no per-tile cross-lane reductions, no runtime
// branching in the hot epilogue).
// ---------------------------------------------------------------------------
__global__ __launch_bounds__(256) void detconb_main(
    const float* __restrict__ p1n, const float* __restrict__ p2n,
    const float* __restrict__ t1n, const float* __restrict__ t2n,
    const int* __restrict__ pind1, const int* __restrict__ pind2,
    const int* __restrict__ tind1, const int* __restrict__ tind2,
    float* __restrict__ partial)
{
    // A panels with padded row stride (260 % 64 == 4 banks shift per row ->
    // b64 fragment loads are bank-conflict free across the wave).
    __shared__ float lA[2][NROI][260];
    __shared__ float redS[8][2][NROI];
    __shared__ float redP[8][2][NROI];
    __shared__ float redC[8][2][NROI];
    __shared__ int   sPind[2][NROI];
    __shared__ int   sTind[2][NROI];

    const int b    = blockIdx.x;
    const int tid  = threadIdx.x;
    const int wave = tid >> 5;
    const int lane = tid & 31;
    const int mloc = lane & 15;   // column within tile (N), and M within half
    const int half = lane >> 4;   // 0: lanes 0-15, 1: lanes 16-31

    // ---- stage A panels (p1[b], p2[b]) into LDS -------------------------
    #pragma unroll
    for (int br = 0; br < 2; ++br) {
        const float* src = (br ? p2n : p1n) + (size_t)b * (NROI * DDIM);
        #pragma unroll
        for (int i = 0; i < 4; ++i) {
            const int flat = (tid + i * 256) * 4;       // 0..4092 step 4
            const int row  = flat >> 8;
            const int col  = flat & 255;
            *(v4f*)&lA[br][row][col] = *(const v4f*)(src + flat);
        }
    }
    // ---- stage indices --------------------------------------------------
    if (tid < 64) {
        const int k = tid >> 4, j = tid & 15;
        const int* srcs[4] = { pind1, pind2, tind1, tind2 };
        const int v = srcs[k][b * NROI + j];
        if (k < 2) sPind[k][j] = v; else sTind[k - 2][j] = v;
    }
    __syncthreads();

    // hoist this lane's mask ids: rows j and j+8 (per half), column mloc
    int pid[2][8];
    #pragma unroll
    for (int br = 0; br < 2; ++br)
        #pragma unroll
        for (int j = 0; j < 8; ++j) pid[br][j] = sPind[br][j + 8 * half];
    const int tiv[2] = { sTind[0][mloc], sTind[1][mloc] };

    // lane-local running state: sum-exp (fixed bias), positive sum/count
    float rs[2][8], rp[2][8], rc[2][8];
    #pragma unroll
    for (int br = 0; br < 2; ++br)
        #pragma unroll
        for (int j = 0; j < 8; ++j) { rs[br][j] = 0.0f; rp[br][j] = 0.0f;
                                      rc[br][j] = 0.0f; }

    const float* A0 = &lA[0][mloc][2 * half];
    const float* A1 = &lA[1][mloc][2 * half];

    #pragma unroll                                 // make `positive` compile-time
    for (int ts = 0; ts < 2; ++ts) {               // ts=0 -> T1, ts=1 -> T2
        const float* T = ts ? t2n : t1n;
        const int tv = tiv[ts];
        for (int i = 0; i < 16; ++i) {             // 2 tiles / iter, 32 per wave
            const int ctA = 2 * wave + 16 * i;     // even tile
            // B fragment bases: tile pair covers 32 contiguous target rows
            const float* BpA = T + ((size_t)(ctA * NROI + mloc)) * DDIM + 2 * half;
            const float* BpB = BpA + (size_t)NROI * DDIM;
            v8f acc[2][2];                         // [tile][branch]
            #pragma unroll
            for (int t = 0; t < 2; ++t)
                #pragma unroll
                for (int br = 0; br < 2; ++br)
                    acc[t][br] = (v8f){0.f,0.f,0.f,0.f,0.f,0.f,0.f,0.f};

            #pragma unroll 4
            for (int kk = 0; kk < 64; ++kk) {
                const v2f bfa = *(const v2f*)(BpA + 4 * kk);
                const v2f bfb = *(const v2f*)(BpB + 4 * kk);
                const v2f a0  = *(const v2f*)(A0 + 4 * kk);
                const v2f a1  = *(const v2f*)(A1 + 4 * kk);
                acc[0][0] = __builtin_amdgcn_wmma_f32_16x16x4_f32(
                        false, a0, false, bfa, (short)0, acc[0][0], false, false);
                acc[0][1] = __builtin_amdgcn_wmma_f32_16x16x4_f32(
                        false, a1, false, bfa, (short)0, acc[0][1], false, false);
                acc[1][0] = __builtin_amdgcn_wmma_f32_16x16x4_f32(
                        false, a0, false, bfb, (short)0, acc[1][0], false, false);
                acc[1][1] = __builtin_amdgcn_wmma_f32_16x16x4_f32(
                        false, a1, false, bfb, (short)0, acc[1][1], false, false);
            }

            // ---- lane-local epilogue --------------------------------------
            // Hot path per element: v_mul + v_exp + v_add. Mask / positive
            // handling only under the wave-uniform `diag` branch (1 tile/32).
            #pragma unroll
            for (int t = 0; t < 2; ++t) {
                const bool diag = ((ctA + t) == b);
                #pragma unroll
                for (int br = 0; br < 2; ++br) {
                    // branch0 (p1): T2 => logits_ab (positives); T1 => logits_aa (mask)
                    // branch1 (p2): T1 => logits_ba (positives); T2 => logits_bb (mask)
                    constexpr bool posTbl[2][2] = {{false, true}, {true, false}};
                    const bool positive = posTbl[br][ts];
                    #pragma unroll
                    for (int j = 0; j < 8; ++j) {
                        float val = acc[t][br][j] * LOGIT_SCALE;
                        if (diag) {                        // wave-uniform, rare
                            const bool match = (pid[br][j] == tv);
                            if (positive) {
                                rp[br][j] += match ? val : 0.0f;   // v_cndmask
                                rc[br][j] += match ? 1.0f : 0.0f;
                            } else {
                                val = match ? val - 1e9f : val;    // -INF mask
                            }
                        }
                        rs[br][j] += __expf(val - LSE_BIAS); // exact: val <= 10
                    }
                }
            }
        }
    }

    // ---- one cross-lane reduction at the very end -----------------------
    #pragma unroll
    for (int br = 0; br < 2; ++br)
        #pragma unroll
        for (int j = 0; j < 8; ++j) {
            float s = rs[br][j], p = rp[br][j], c = rc[br][j];
            #pragma unroll
            for (int m = 1; m < 16; m <<= 1) {
                s += __shfl_xor(s, m, 32);
                p += __shfl_xor(p, m, 32);
                c += __shfl_xor(c, m, 32);
            }
            if (mloc == 0) {
                const int row = j + 8 * half;
                redS[wave][br][row] = s;
                redP[wave][br][row] = p;
                redC[wave][br][row] = c;
            }
        }
    __syncthreads();

    if (tid < 32) {
        const int br = tid >> 4, row = tid & 15;
        float s = 0.0f, ps = 0.0f, pc = 0.0f;
        #pragma unroll
        for (int w = 0; w < 8; ++w) {
            s  += redS[w][br][row];
            ps += redP[w][br][row];
            pc += redC[w][br][row];
        }
        const float lse = LSE_BIAS + logf(s);
        const int* pi = sPind[br];
        const int myid = pi[row];
        int area = 0;
        #pragma unroll
        for (int mm = 0; mm < NROI; ++mm) area += (myid == pi[mm]);
        float ce = 0.0f;
        if (pc > 0.001f) {
            const float w0 = 1.0f / (float)area;
            ce = -w0 * (ps / pc - lse);
        }
        // 32 ce values live in wave 0 -> full-wave reduction, deterministic
        #pragma unroll
        for (int m2 = 1; m2 < 32; m2 <<= 1) ce += __shfl_xor(ce, m2, 32);
        if (tid == 0) partial[blockIdx.x] = ce;
    }
}

// ---------------------------------------------------------------------------
// Kernel 3: deterministic final reduction of 256 block partials.
// ---------------------------------------------------------------------------
__global__ __launch_bounds__(256) void detconb_reduce(
    const float* __restrict__ partial, float* __restrict__ out)
{
    __shared__ float s[256];
    s[threadIdx.x] = partial[threadIdx.x];
    __syncthreads();
    for (int st = 128; st > 0; st >>= 1) {
        if (threadIdx.x < st) s[threadIdx.x] += s[threadIdx.x + st];
        __syncthreads();
    }
    if (threadIdx.x == 0) out[0] = s[0] / (float)ROWS;  // mean(ce0)+mean(ce1)
}

// ---------------------------------------------------------------------------
extern "C" void kernel_launch(void* const* d_in, const int* in_sizes, int n_in,
                              void* d_out, int out_size, void* d_ws, size_t ws_size,
                              hipStream_t stream) {
    const float* p1 = (const float*)d_in[0];
    const float* p2 = (const float*)d_in[1];
    const float* t1 = (const float*)d_in[2];
    const float* t2 = (const float*)d_in[3];
    const int* pind1 = (const int*)d_in[4];
    const int* pind2 = (const int*)d_in[5];
    const int* tind1 = (const int*)d_in[6];
    const int* tind2 = (const int*)d_in[7];

    float* ws  = (float*)d_ws;
    float* p1n = ws;
    float* p2n = ws + 1 * (size_t)TENSOR_ELEMS;
    float* t1n = ws + 2 * (size_t)TENSOR_ELEMS;
    float* t2n = ws + 3 * (size_t)TENSOR_ELEMS;
    float* partial = ws + 4 * (size_t)TENSOR_ELEMS;  // 256 floats

    // 16384 rows, one wave each, 8 waves per block -> 2048 blocks
    detconb_normalize<<<2048, 256, 0, stream>>>(p1, p2, t1, t2, ws);
    detconb_main<<<BATCH, 256, 0, stream>>>(p1n, p2n, t1n, t2n,
                                            pind1, pind2, tind1, tind2, partial);
    detconb_reduce<<<1, 256, 0, stream>>>(partial, (float*)d_out);
}